// GuidedFilter_32478542692834
// MI455X (gfx1250) — compile-verified
//
#include <hip/hip_runtime.h>

// Guided filter, r=8, eps=1e-4, on (3, 4096, 4096) fp32.
// - Box sums are band-matrix multiplies on the CDNA5 matrix pipe via
//   V_WMMA_F32_16X16X4_F32 (fp32 end-to-end; covariance subtraction cannot
//   tolerate f16 rounding of the means).
// - Halo tile load uses CDNA5 async global->LDS (GLOBAL_LOAD_ASYNC_TO_LDS_B128,
//   ASYNCcnt) for interior blocks; border blocks use a guarded zero-padded path.
// - Wave-task loops are driven by a readfirstlane'd wave id so control flow is
//   scalar and EXEC is all-1s at every WMMA (ISA requirement).
//
// Memory-bound problem: ~604 MB essential HBM traffic ~= 26 us @ 23.3 TB/s.
// 48x48 halo re-reads between neighboring 32x32 tiles are absorbed by the
// 192 MB L2 (a full tile-row band of halo is ~3 MB).

#define RADIUS 8
#define TILE   32
#define HALO   48          // TILE + 2*RADIUS
#define IMG_H  4096
#define IMG_W  4096
#define EPS_F  1e-4f

typedef __attribute__((ext_vector_type(2))) float v2f;
typedef __attribute__((ext_vector_type(8))) float v8f;

__global__ __launch_bounds__(256)
void guided_filter_wmma_kernel(const float* __restrict__ img_i,
                               const float* __restrict__ img_p,
                               float* __restrict__ out)
{
    // LDS word layout (15360 floats = 60 KB):
    //   [0)     i  halo tile  48x48   (2304)  -- live until epilogue
    //   [2304)  p  halo tile  48x48   (2304)
    //   [4608)  pi halo tile  48x48   (2304)  -- dead after vertical pass
    //   [6912)  ii halo tile  48x48   (2304)  -- dead after vertical pass
    //   [9216)  V[q] 32x48, q=0..3    (6144)
    //   sums S[q] 32x32 (4096 words) aliased at [4608) over pi/ii.
    __shared__ float lds[15360];
    float* vbuf = lds + 9216;
    float* sbuf = lds + 4608;

    const int tid  = threadIdx.x;
    const int lane = tid & 31;
    const int wv   = __builtin_amdgcn_readfirstlane(tid >> 5);  // scalar wave id
    const int half = lane >> 4;      // which 16-lane half
    const int m16  = lane & 15;      // M index (A) / N index (B) / col (C,D)

    const int bx = blockIdx.x, by = blockIdx.y, ch = blockIdx.z;
    const long chbase = (long)ch * IMG_H * IMG_W;

    // ---- Phase 1: halo load of i and p ----
    const int gr0 = by * TILE - RADIUS;
    const int gc0 = bx * TILE - RADIUS;
    const bool interior = (gr0 >= 0) && (gc0 >= 0) &&
                          (gr0 + HALO <= IMG_H) && (gc0 + HALO <= IMG_W);
    if (interior) {
        // Async global->LDS, 16B granules. Halo row = 48 floats = 12 x b128,
        // all 16B aligned (row start byte offset = (32*bx - 8)*4 % 16 == 0).
        const float* src_i = img_i + chbase + (long)gr0 * IMG_W + gc0;
        const float* src_p = img_p + chbase + (long)gr0 * IMG_W + gc0;
        const unsigned lds_i = (unsigned)(uintptr_t)&lds[0];     // addr[31:0] == LDS offset
        const unsigned lds_p = (unsigned)(uintptr_t)&lds[2304];
        for (int it = 0; it < 3; ++it) {
            int idx4 = tid + it * 256;            // 0..575 (576 = 2304/4)
            if (idx4 < 576) {
                int r  = idx4 / 12;
                int c4 = idx4 % 12;
                unsigned long long ga_i =
                    (unsigned long long)(uintptr_t)(src_i + (long)r * IMG_W + c4 * 4);
                unsigned long long ga_p =
                    (unsigned long long)(uintptr_t)(src_p + (long)r * IMG_W + c4 * 4);
                unsigned ldo_i = lds_i + (unsigned)idx4 * 16u;
                unsigned ldo_p = lds_p + (unsigned)idx4 * 16u;
                asm volatile("global_load_async_to_lds_b128 %0, %1, off"
                             :: "v"(ldo_i), "v"(ga_i) : "memory");
                asm volatile("global_load_async_to_lds_b128 %0, %1, off"
                             :: "v"(ldo_p), "v"(ga_p) : "memory");
            }
        }
        asm volatile("s_wait_asynccnt 0x0" ::: "memory");
    } else {
        // Border blocks: guarded loads with zero padding (truncated window).
        for (int it = 0; it < 9; ++it) {
            int idx = tid + it * 256;             // 0..2303
            int r = idx / HALO, c = idx % HALO;
            int gr = gr0 + r;
            int gc = gc0 + c;
            float vi = 0.0f, vp = 0.0f;
            if (gr >= 0 && gr < IMG_H && gc >= 0 && gc < IMG_W) {
                long g = chbase + (long)gr * IMG_W + gc;
                vi = img_i[g];
                vp = img_p[g];
            }
            lds[idx]        = vi;
            lds[2304 + idx] = vp;
        }
    }
    __syncthreads();

    // ---- Phase 2: product tiles p*i and i*i ----
    for (int it = 0; it < 9; ++it) {
        int idx = tid + it * 256;
        float vi = lds[idx], vp = lds[2304 + idx];
        lds[4608 + idx] = vp * vi;
        lds[6912 + idx] = vi * vi;
    }
    __syncthreads();

    // ---- Phase 3: vertical 17-tap sum via WMMA (24 wave-tasks, scalar loop) ----
    // V(16x16) = A_band(16x32) @ T(32x16): A[m][k] = 1 iff m <= k <= m+16,
    // B[k][n] = T[tr*16 + k][col].
    for (int t = wv; t < 24; t += 8) {
        int q    = t / 6;
        int tile = t % 6;
        int tr = tile / 3, tc = tile % 3;
        const float* tq = lds + q * 2304;
        int col = tc * 16 + m16;                  // halo col 0..47
        v8f acc = {};
#pragma unroll
        for (int j = 0; j < 8; ++j) {
            int k0 = 4 * j + half * 2;            // K context index 0..31
            v2f a, b;
            a.x = (k0     >= m16 && k0     <= m16 + 16) ? 1.0f : 0.0f;
            a.y = (k0 + 1 >= m16 && k0 + 1 <= m16 + 16) ? 1.0f : 0.0f;
            int rr = tr * 16 + k0;                // halo row
            b.x = tq[rr * HALO + col];
            b.y = tq[(rr + 1) * HALO + col];
            acc = __builtin_amdgcn_wmma_f32_16x16x4_f32(
                      false, a, false, b, (short)0, acc, false, false);
        }
        union { v8f v; float f[8]; } u; u.v = acc;
        float* vq = vbuf + q * 1536;
#pragma unroll
        for (int vv = 0; vv < 8; ++vv) {
            int row = tr * 16 + vv + half * 8;    // output row 0..31
            vq[row * HALO + col] = u.f[vv];
        }
    }
    __syncthreads();

    // ---- Phase 4: horizontal 17-tap sum via WMMA (16 wave-tasks, scalar loop) ----
    // S(16x16) = V(16x32) @ B_band(32x16): A[m][k] = V[row][c0+k] (data),
    // B[k][n] = 1 iff n <= k <= n+16.
    for (int t = wv; t < 16; t += 8) {
        int q    = t / 4;
        int tile = t % 4;
        int tr = tile / 2, tc = tile % 2;
        const float* vq = vbuf + q * 1536;
        int row = tr * 16 + m16;                  // output row 0..31
        v8f acc = {};
#pragma unroll
        for (int j = 0; j < 8; ++j) {
            int k0 = 4 * j + half * 2;
            v2f a, b;
            a.x = vq[row * HALO + tc * 16 + k0];
            a.y = vq[row * HALO + tc * 16 + k0 + 1];
            b.x = (k0     >= m16 && k0     <= m16 + 16) ? 1.0f : 0.0f;
            b.y = (k0 + 1 >= m16 && k0 + 1 <= m16 + 16) ? 1.0f : 0.0f;
            acc = __builtin_amdgcn_wmma_f32_16x16x4_f32(
                      false, a, false, b, (short)0, acc, false, false);
        }
        union { v8f v; float f[8]; } u; u.v = acc;
        float* sq = sbuf + q * 1024;
#pragma unroll
        for (int vv = 0; vv < 8; ++vv) {
            int orow = tr * 16 + vv + half * 8;
            int ocol = tc * 16 + m16;
            sq[orow * TILE + ocol] = u.f[vv];
        }
    }
    __syncthreads();

    // ---- Phase 5: epilogue (4 pixels/thread, coalesced 128B stores/wave) ----
    for (int kk = 0; kk < 4; ++kk) {
        int pid = tid + kk * 256;                 // 0..1023 == row*32 + col
        int row = pid >> 5;
        int col = pid & 31;
        float s_i  = sbuf[0 * 1024 + pid];
        float s_p  = sbuf[1 * 1024 + pid];
        float s_pi = sbuf[2 * 1024 + pid];
        float s_ii = sbuf[3 * 1024 + pid];

        int gr = by * TILE + row;
        int gc = bx * TILE + col;
        // border-truncated window pixel count (separable, analytic)
        int rlo = gr - RADIUS; if (rlo < 0) rlo = 0;
        int rhi = gr + RADIUS; if (rhi > IMG_H - 1) rhi = IMG_H - 1;
        int clo = gc - RADIUS; if (clo < 0) clo = 0;
        int chi = gc + RADIUS; if (chi > IMG_W - 1) chi = IMG_W - 1;
        float Ninv = 1.0f / (float)((rhi - rlo + 1) * (chi - clo + 1));

        float mean_i  = s_i  * Ninv;
        float mean_p  = s_p  * Ninv;
        float mean_pi = s_pi * Ninv;
        float mean_ii = s_ii * Ninv;
        float cov_ip  = mean_pi - mean_p * mean_i;
        float cov_ii  = mean_ii - mean_i * mean_i;
        float a = cov_ip / (cov_ii + EPS_F);
        float b = mean_p - a * mean_i;

        float iv = lds[(row + RADIUS) * HALO + (col + RADIUS)];  // i tile still live
        float qv = a * iv + b;
        qv = qv < 0.0f ? 0.0f : (qv > 1.0f ? 1.0f : qv);
        out[chbase + (long)gr * IMG_W + gc] = qv;
    }
}

extern "C" void kernel_launch(void* const* d_in, const int* in_sizes, int n_in,
                              void* d_out, int out_size, void* d_ws, size_t ws_size,
                              hipStream_t stream) {
    const float* i = (const float*)d_in[0];
    const float* p = (const float*)d_in[1];
    float* out = (float*)d_out;
    dim3 grid(IMG_W / TILE, IMG_H / TILE, 3);
    guided_filter_wmma_kernel<<<grid, 256, 0, stream>>>(i, p, out);
}